// GlobalAvgPool_30073361006684
// MI455X (gfx1250) — compile-verified
//
#include <hip/hip_runtime.h>
#include <hip/hip_bf16.h>

typedef __attribute__((ext_vector_type(2))) float v2f;
typedef __attribute__((ext_vector_type(8))) float v8f;

#define GAP_HW     3136   // 56*56 spatial elements per channel
#define GAP_SLICE  392    // spatial elements per wave (3136 / 8 waves)
#define GAP_ITERS  49     // 392 / 8 floats consumed per iteration
#define GAP_CH     49152  // 64*768 output channels

__global__ __launch_bounds__(256)
void GlobalAvgPool_wmma_kernel(const float* __restrict__ x, float* __restrict__ out) {
    const int g    = blockIdx.x;        // group of 16 channels
    const int tid  = threadIdx.x;
    const int wave = tid >> 5;          // 0..7 (wave32)
    const int lane = tid & 31;
    const int m    = lane & 15;         // matrix row = channel within group
    const int hi   = lane >> 4;         // 0: K0/K1 half, 1: K2/K3 half

    // Each lane streams one channel: lo lanes read floats [s..s+3], hi lanes [s+4..s+7].
    const float* base = x + (size_t)(g * 16 + m) * GAP_HW
                          + (size_t)(wave * GAP_SLICE) + hi * 4;

    const v2f ones = {1.0f, 1.0f};      // B matrix (4x16) of ones -> D = row-sums of A
    v8f c0 = {};
    v8f c1 = {};

    #pragma unroll 7
    for (int i = 0; i < GAP_ITERS; ++i) {
        const float4 v = *(const float4*)(base + i * 8);
        v2f a0 = {v.x, v.y};            // lo: K0,K1 = f0,f1 | hi: K2,K3 = f4,f5
        v2f a1 = {v.z, v.w};            // lo: K0,K1 = f2,f3 | hi: K2,K3 = f6,f7
        c0 = __builtin_amdgcn_wmma_f32_16x16x4_f32(false, a0, false, ones,
                                                   (short)0, c0, false, false);
        c1 = __builtin_amdgcn_wmma_f32_16x16x4_f32(false, a1, false, ones,
                                                   (short)0, c1, false, false);
    }

    // Merge the two accumulators; every column of D holds the same row-sum.
    // Row r (0..7)  sum: c[r] at lanes 0..15
    // Row 8+r       sum: c[r] at lanes 16..31
    v8f c;
    #pragma unroll
    for (int r = 0; r < 8; ++r) c[r] = c0[r] + c1[r];

    __shared__ float lds[8 * 16];       // [wave][row]
    if (lane == 0 || lane == 16) {
        float* p = &lds[wave * 16 + hi * 8];
        #pragma unroll
        for (int r = 0; r < 8; ++r) p[r] = c[r];
    }
    __syncthreads();

    if (tid < 16) {
        float s = 0.0f;
        #pragma unroll
        for (int w = 0; w < 8; ++w) s += lds[w * 16 + tid];
        out[g * 16 + tid] = s * (1.0f / (float)GAP_HW);
    }
}

extern "C" void kernel_launch(void* const* d_in, const int* in_sizes, int n_in,
                              void* d_out, int out_size, void* d_ws, size_t ws_size,
                              hipStream_t stream) {
    (void)in_sizes; (void)n_in; (void)d_ws; (void)ws_size; (void)out_size;
    const float* x  = (const float*)d_in[0];
    float* out      = (float*)d_out;
    const int groups = GAP_CH / 16;     // 3072 workgroups of 256 threads
    GlobalAvgPool_wmma_kernel<<<groups, 256, 0, stream>>>(x, out);
}